// SmileMoELinear_15109694947873
// MI455X (gfx1250) — compile-verified
//
#include <hip/hip_runtime.h>
#include <math.h>

typedef __attribute__((ext_vector_type(2))) float v2f;
typedef __attribute__((ext_vector_type(8))) float v8f;

#define T_TOK   8192
#define IN_F    2048
#define OUT_F   2048
#define NEXP    8
#define GKDIM   16
#define RANK    32
#define KEXT    272   /* 256 (E*K) + 8 (expert bias) + 8 pad -> multiple of BK */

#define BM 64
#define BN 128
#define BK 16
#define SK 20         /* LDS row stride: BK + 4 pad (16B-aligned float4 stores, conflict-free b64 reads) */

// ---------------------------------------------------------------------------
// Kernel 1: router. One wave32 per token. g = x @ gate_w^T -> per-expert L2
// norm -> top-2 of logits -> normalized weights into Aext cols 256..271
// (cols 264..271 are the zero K-padding).
// ---------------------------------------------------------------------------
__global__ __launch_bounds__(256) void router_kernel(
    const float* __restrict__ x, const float* __restrict__ gw,
    float* __restrict__ Aext)
{
    const int lane = threadIdx.x & 31;
    const int wv   = threadIdx.x >> 5;
    const int t    = blockIdx.x * 8 + wv;
    const float* __restrict__ xr = x + (size_t)t * IN_F;

    float lg[NEXP];
#pragma unroll
    for (int e = 0; e < NEXP; ++e) {
        float ss = 0.f;
        for (int j = 0; j < GKDIM; ++j) {
            const float* __restrict__ gr = gw + (size_t)(e * GKDIM + j) * IN_F;
            float p = 0.f;
            for (int k = lane; k < IN_F; k += 32)
                p += xr[k] * gr[k];
#pragma unroll
            for (int off = 16; off >= 1; off >>= 1)
                p += __shfl_xor(p, off, 32);
            ss += p * p;
        }
        lg[e] = sqrtf(ss);
    }
    // top-2 (first index wins ties, matching lax.top_k)
    int i1 = 0;
#pragma unroll
    for (int e = 1; e < NEXP; ++e) if (lg[e] > lg[i1]) i1 = e;
    int i2 = (i1 == 0) ? 1 : 0;
#pragma unroll
    for (int e = 0; e < NEXP; ++e) if (e != i1 && lg[e] > lg[i2]) i2 = e;
    const float e2 = __expf(lg[i2] - lg[i1]);
    const float w1 = 1.f / (1.f + e2);
    const float w2 = e2 / (1.f + e2);

    if (lane < 16) {
        float v = 0.f;
        if (lane == i1) v = w1;
        else if (lane == i2) v = w2;   // lanes 8..15 stay 0 -> K padding
        Aext[(size_t)t * KEXT + 256 + lane] = v;
    }
}

// ---------------------------------------------------------------------------
// Kernel 2: build Mext[OUT_F x KEXT]:
//   col r < 256      : u[e][o][k]   (e = r/32, k = r%32)
//   col 256..263     : expert_bias[r-256][o]
//   col 264..271     : 0
// ---------------------------------------------------------------------------
__global__ void build_mext_kernel(const float* __restrict__ u,
                                  const float* __restrict__ eb,
                                  float* __restrict__ Mext)
{
    const int idx = blockIdx.x * 256 + threadIdx.x;
    if (idx >= OUT_F * KEXT) return;
    const int o = idx / KEXT;
    const int r = idx % KEXT;
    float v;
    if (r < 256) {
        const int e = r >> 5, k = r & 31;
        v = u[(size_t)e * OUT_F * RANK + (size_t)o * RANK + k];
    } else if (r < 264) {
        v = eb[(size_t)(r - 256) * OUT_F + o];
    } else {
        v = 0.f;
    }
    Mext[idx] = v;
}

// ---------------------------------------------------------------------------
// Shared tiled fp32-WMMA GEMM phase: acc += A[M x K] @ Mrow[N x K]^T
// Block tile 64x128, 8 waves (2x4), wave tile 32x32 (4 x v8f accumulators).
// ---------------------------------------------------------------------------
__device__ __forceinline__ void gemm_phase(
    float* __restrict__ As, float* __restrict__ Bs, v8f (&acc)[2][2],
    const float* __restrict__ A, int ldA,
    const float* __restrict__ Bm, int ldB, int K,
    int m0, int n0, int tid, int lane, int wm, int wn)
{
    const int ar = tid >> 2, ac = (tid & 3) << 2;   // A tile: 64 rows x 16
    const int br = tid >> 1, bc = (tid & 1) << 3;   // B tile: 128 rows x 16
    const int hi  = (lane >> 4) << 1;               // half-wave K split
    const int rA0 = wm * 32 + (lane & 15);
    const int rB0 = wn * 32 + (lane & 15);

    for (int k0 = 0; k0 < K; k0 += BK) {
        __syncthreads();
        const float4 av  = *(const float4*)(A  + (size_t)(m0 + ar) * ldA + k0 + ac);
        const float4 bv0 = *(const float4*)(Bm + (size_t)(n0 + br) * ldB + k0 + bc);
        const float4 bv1 = *(const float4*)(Bm + (size_t)(n0 + br) * ldB + k0 + bc + 4);
        *(float4*)(As + ar * SK + ac)     = av;
        *(float4*)(Bs + br * SK + bc)     = bv0;
        *(float4*)(Bs + br * SK + bc + 4) = bv1;
        __syncthreads();
#pragma unroll
        for (int kk = 0; kk < BK; kk += 4) {
            const v2f a0 = *(const v2f*)(As + (rA0)      * SK + kk + hi);
            const v2f a1 = *(const v2f*)(As + (rA0 + 16) * SK + kk + hi);
            const v2f b0 = *(const v2f*)(Bs + (rB0)      * SK + kk + hi);
            const v2f b1 = *(const v2f*)(Bs + (rB0 + 16) * SK + kk + hi);
            acc[0][0] = __builtin_amdgcn_wmma_f32_16x16x4_f32(false, a0, false, b0, (short)0, acc[0][0], false, false);
            acc[0][1] = __builtin_amdgcn_wmma_f32_16x16x4_f32(false, a0, false, b1, (short)0, acc[0][1], false, false);
            acc[1][0] = __builtin_amdgcn_wmma_f32_16x16x4_f32(false, a1, false, b0, (short)0, acc[1][0], false, false);
            acc[1][1] = __builtin_amdgcn_wmma_f32_16x16x4_f32(false, a1, false, b1, (short)0, acc[1][1], false, false);
        }
    }
}

// ---------------------------------------------------------------------------
// Kernel 3: zw = (x @ svh_all^T) * w_dense  -> Aext cols 0..255
// svh [E,K,IN] flattens exactly to svh_all [256, IN] with row = e*32+k.
// ---------------------------------------------------------------------------
__global__ __launch_bounds__(256) void zw_kernel(
    const float* __restrict__ x, const float* __restrict__ svh,
    float* __restrict__ Aext)
{
    __shared__ float As[BM * SK];
    __shared__ float Bs[BN * SK];
    const v8f zero = {0.f, 0.f, 0.f, 0.f, 0.f, 0.f, 0.f, 0.f};
    v8f acc[2][2] = {{zero, zero}, {zero, zero}};

    const int tid = threadIdx.x, lane = tid & 31, wv = tid >> 5;
    const int wm = wv & 1, wn = wv >> 1;
    const int m0 = blockIdx.x * BM, n0 = blockIdx.y * BN;

    gemm_phase(As, Bs, acc, x, IN_F, svh, IN_F, IN_F, m0, n0, tid, lane, wm, wn);

    const int colBase = n0 + wn * 32 + (lane & 15);
    const int rowBase = m0 + wm * 32 + ((lane >> 4) << 3);
#pragma unroll
    for (int j = 0; j < 2; ++j) {
        const int col = colBase + j * 16;        // r index, 0..255
        const int e   = col >> 5;
#pragma unroll
        for (int i = 0; i < 2; ++i) {
#pragma unroll
            for (int v = 0; v < 8; ++v) {
                const int row = rowBase + i * 16 + v;
                const float wd = Aext[(size_t)row * KEXT + 256 + e];
                Aext[(size_t)row * KEXT + col] = acc[i][j][v] * wd;
            }
        }
    }
}

// ---------------------------------------------------------------------------
// Kernel 4: out = x @ W^T  +  Aext @ Mext^T  +  b
// Phase 2 folds in: low-rank experts (zw @ U2) + w_dense @ expert_bias.
// ---------------------------------------------------------------------------
__global__ __launch_bounds__(256) void fused_gemm_kernel(
    const float* __restrict__ x, const float* __restrict__ W,
    const float* __restrict__ b, const float* __restrict__ Aext,
    const float* __restrict__ Mext, float* __restrict__ out)
{
    __shared__ float As[BM * SK];
    __shared__ float Bs[BN * SK];
    const v8f zero = {0.f, 0.f, 0.f, 0.f, 0.f, 0.f, 0.f, 0.f};
    v8f acc[2][2] = {{zero, zero}, {zero, zero}};

    const int tid = threadIdx.x, lane = tid & 31, wv = tid >> 5;
    const int wm = wv & 1, wn = wv >> 1;
    const int m0 = blockIdx.x * BM, n0 = blockIdx.y * BN;

    gemm_phase(As, Bs, acc, x,    IN_F, W,    IN_F, IN_F, m0, n0, tid, lane, wm, wn);
    gemm_phase(As, Bs, acc, Aext, KEXT, Mext, KEXT, KEXT, m0, n0, tid, lane, wm, wn);

    const int colBase = n0 + wn * 32 + (lane & 15);
    const int rowBase = m0 + wm * 32 + ((lane >> 4) << 3);
#pragma unroll
    for (int j = 0; j < 2; ++j) {
        const int col  = colBase + j * 16;
        const float bs = b[col];
#pragma unroll
        for (int i = 0; i < 2; ++i) {
#pragma unroll
            for (int v = 0; v < 8; ++v) {
                const int row = rowBase + i * 16 + v;
                out[(size_t)row * OUT_F + col] = acc[i][j][v] + bs;
            }
        }
    }
}

// ---------------------------------------------------------------------------
extern "C" void kernel_launch(void* const* d_in, const int* in_sizes, int n_in,
                              void* d_out, int out_size, void* d_ws, size_t ws_size,
                              hipStream_t stream) {
    (void)in_sizes; (void)n_in; (void)out_size; (void)ws_size;
    const float* x   = (const float*)d_in[0];  // [4,2048,2048]
    const float* W   = (const float*)d_in[1];  // [2048,2048]
    const float* b   = (const float*)d_in[2];  // [2048]
    const float* gw  = (const float*)d_in[3];  // [128,2048]
    const float* u   = (const float*)d_in[4];  // [8,2048,32]
    const float* svh = (const float*)d_in[5];  // [8,32,2048]
    const float* eb  = (const float*)d_in[6];  // [8,2048]
    // d_in[7] = top_k (==2, hardcoded)
    float* out = (float*)d_out;

    float* Aext = (float*)d_ws;                       // T_TOK * KEXT floats
    float* Mext = Aext + (size_t)T_TOK * KEXT;        // OUT_F * KEXT floats

    router_kernel<<<T_TOK / 8, 256, 0, stream>>>(x, gw, Aext);
    build_mext_kernel<<<(OUT_F * KEXT + 255) / 256, 256, 0, stream>>>(u, eb, Mext);
    zw_kernel<<<dim3(T_TOK / BM, 256 / BN), 256, 0, stream>>>(x, svh, Aext);
    fused_gemm_kernel<<<dim3(T_TOK / BM, OUT_F / BN), 256, 0, stream>>>(x, W, b, Aext, Mext, out);
}